// GraphVAE_68092411511099
// MI455X (gfx1250) — compile-verified
//
#include <hip/hip_runtime.h>
#include <math.h>

typedef __attribute__((ext_vector_type(2))) float v2f;
typedef __attribute__((ext_vector_type(8))) float v8f;

#define NN   2048
#define NE   65536
#define INC  128
#define HIDD 32
#define LATD 16

// ---------------- utility ----------------
__global__ void k_fill(float* __restrict__ p, float v, int n) {
  int i = blockIdx.x * blockDim.x + threadIdx.x;
  if (i < n) p[i] = v;
}

// deg[dst] += 1 over edges (deg pre-filled with 1.0 for self-loop)
__global__ void k_deg_scatter(const long long* __restrict__ ei, float* __restrict__ deg) {
  int e = blockIdx.x * blockDim.x + threadIdx.x;
  if (e < NE) atomicAdd(&deg[(int)ei[NE + e]], 1.0f);
}

__global__ void k_dinv(const float* __restrict__ deg, float* __restrict__ dinv, int n) {
  int i = blockIdx.x * blockDim.x + threadIdx.x;
  if (i < n) dinv[i] = rsqrtf(deg[i]);
}

// ---------------- WMMA GEMMs (f32 16x16x4, wave32: one wave per 16x16 C tile) ----------------
// h1[2048x32] = x[2048x128] @ W1[128x32]
__global__ __launch_bounds__(256) void k_gemm1(const float* __restrict__ x,
                                               const float* __restrict__ W1,
                                               float* __restrict__ h1) {
  int wid  = (blockIdx.x * 256 + threadIdx.x) >> 5;   // 256 waves total
  int lane = threadIdx.x & 31;
  int tileN = wid & 1;          // 2 N-tiles
  int tileM = wid >> 1;         // 128 M-tiles
  int m   = tileM * 16 + (lane & 15);
  int kk  = (lane >> 4) << 1;   // 0 or 2
  int col = tileN * 16 + (lane & 15);
  v8f c = {};
#pragma unroll
  for (int k = 0; k < INC; k += 4) {
    v2f a = *(const v2f*)(x + m * INC + k + kk);
    v2f b;
    b.x = W1[(k + kk) * HIDD + col];
    b.y = W1[(k + kk + 1) * HIDD + col];
    c = __builtin_amdgcn_wmma_f32_16x16x4_f32(false, a, false, b, (short)0, c, false, false);
  }
  int rowbase = tileM * 16 + ((lane >> 4) << 3);
#pragma unroll
  for (int v = 0; v < 8; ++v)
    h1[(rowbase + v) * HIDD + col] = c[v];
}

// mupre = h @ W_mu, lvpre = h @ W_lv   (h:2048x32, W:32x16) fused, shared A frags
__global__ __launch_bounds__(256) void k_gemm2(const float* __restrict__ h,
                                               const float* __restrict__ Wmu,
                                               const float* __restrict__ Wlv,
                                               float* __restrict__ mupre,
                                               float* __restrict__ lvpre) {
  int wid  = (blockIdx.x * 256 + threadIdx.x) >> 5;   // 128 waves
  int lane = threadIdx.x & 31;
  int m   = wid * 16 + (lane & 15);
  int kk  = (lane >> 4) << 1;
  int col = lane & 15;
  v8f cm = {}, cl = {};
#pragma unroll
  for (int k = 0; k < HIDD; k += 4) {
    v2f a = *(const v2f*)(h + m * HIDD + k + kk);
    v2f bm, bl;
    bm.x = Wmu[(k + kk) * LATD + col];
    bm.y = Wmu[(k + kk + 1) * LATD + col];
    bl.x = Wlv[(k + kk) * LATD + col];
    bl.y = Wlv[(k + kk + 1) * LATD + col];
    cm = __builtin_amdgcn_wmma_f32_16x16x4_f32(false, a, false, bm, (short)0, cm, false, false);
    cl = __builtin_amdgcn_wmma_f32_16x16x4_f32(false, a, false, bl, (short)0, cl, false, false);
  }
  int rowbase = wid * 16 + ((lane >> 4) << 3);
#pragma unroll
  for (int v = 0; v < 8; ++v) {
    mupre[(rowbase + v) * LATD + col] = cm[v];
    lvpre[(rowbase + v) * LATD + col] = cl[v];
  }
}

// hi = z @ dW1[0:16,:] + db1 (folded), hj = z @ dW1[16:32,:]   (z:2048x16)
__global__ __launch_bounds__(256) void k_gemm3(const float* __restrict__ z,
                                               const float* __restrict__ dW1,
                                               const float* __restrict__ db1,
                                               float* __restrict__ hi,
                                               float* __restrict__ hj) {
  int wid  = (blockIdx.x * 256 + threadIdx.x) >> 5;   // 128 waves
  int lane = threadIdx.x & 31;
  int m   = wid * 16 + (lane & 15);
  int kk  = (lane >> 4) << 1;
  int col = lane & 15;
  v8f ci = {}, cj = {};
#pragma unroll
  for (int k = 0; k < LATD; k += 4) {
    v2f a = *(const v2f*)(z + m * LATD + k + kk);
    v2f bi, bj;
    bi.x = dW1[(k + kk) * LATD + col];
    bi.y = dW1[(k + kk + 1) * LATD + col];
    bj.x = dW1[(LATD + k + kk) * LATD + col];
    bj.y = dW1[(LATD + k + kk + 1) * LATD + col];
    ci = __builtin_amdgcn_wmma_f32_16x16x4_f32(false, a, false, bi, (short)0, ci, false, false);
    cj = __builtin_amdgcn_wmma_f32_16x16x4_f32(false, a, false, bj, (short)0, cj, false, false);
  }
  float bias = db1[col];
  int rowbase = wid * 16 + ((lane >> 4) << 3);
#pragma unroll
  for (int v = 0; v < 8; ++v) {
    hi[(rowbase + v) * LATD + col] = ci[v] + bias;   // db1 folded into hi
    hj[(rowbase + v) * LATD + col] = cj[v];
  }
}

// ---------------- scatter aggregation (GCN message passing) ----------------
__global__ void k_agg1(const long long* __restrict__ ei, const float* __restrict__ dinv,
                       const float* __restrict__ h1, float* __restrict__ agg1) {
  int idx = blockIdx.x * blockDim.x + threadIdx.x;
  if (idx >= NE * HIDD) return;
  int e = idx >> 5, f = idx & 31;
  int s = (int)ei[e], d = (int)ei[NE + e];
  float nrm = dinv[s] * dinv[d];
  atomicAdd(&agg1[d * HIDD + f], nrm * h1[s * HIDD + f]);
}

__global__ void k_relu1(const float* __restrict__ agg1, const float* __restrict__ h1,
                        const float* __restrict__ dinv, const float* __restrict__ b1,
                        float* __restrict__ h) {
  int idx = blockIdx.x * blockDim.x + threadIdx.x;
  if (idx >= NN * HIDD) return;
  int i = idx >> 5, f = idx & 31;
  float di = dinv[i];
  h[idx] = fmaxf(agg1[idx] + di * di * h1[idx] + b1[f], 0.0f);  // self-loop term added here
}

__global__ void k_agg2(const long long* __restrict__ ei, const float* __restrict__ dinv,
                       const float* __restrict__ mupre, const float* __restrict__ lvpre,
                       float* __restrict__ aggmu, float* __restrict__ agglv) {
  int idx = blockIdx.x * blockDim.x + threadIdx.x;
  if (idx >= NE * LATD) return;
  int e = idx >> 4, f = idx & 15;
  int s = (int)ei[e], d = (int)ei[NE + e];
  float nrm = dinv[s] * dinv[d];
  atomicAdd(&aggmu[d * LATD + f], nrm * mupre[s * LATD + f]);
  atomicAdd(&agglv[d * LATD + f], nrm * lvpre[s * LATD + f]);
}

// mu/logvar finalize (self loop + bias) + reparameterize
__global__ void k_finish2(const float* __restrict__ aggmu, const float* __restrict__ agglv,
                          const float* __restrict__ mupre, const float* __restrict__ lvpre,
                          const float* __restrict__ dinv, const float* __restrict__ bmu,
                          const float* __restrict__ blv, const float* __restrict__ eps,
                          float* __restrict__ out_mu, float* __restrict__ out_lv,
                          float* __restrict__ z) {
  int idx = blockIdx.x * blockDim.x + threadIdx.x;
  if (idx >= NN * LATD) return;
  int i = idx >> 4, f = idx & 15;
  float di2 = dinv[i] * dinv[i];
  float m  = aggmu[idx] + di2 * mupre[idx] + bmu[f];
  float lv = agglv[idx] + di2 * lvpre[idx] + blv[f];
  out_mu[idx] = m;
  out_lv[idx] = lv;
  z[idx] = m + eps[idx] * expf(0.5f * lv);
}

// ---------------- pairwise decoder (LDS-tiled, 4 rows x 64 cols per block) ----------------
__global__ __launch_bounds__(256) void k_decoder(const float* __restrict__ hi,
                                                 const float* __restrict__ hj,
                                                 const float* __restrict__ dW2,
                                                 const float* __restrict__ db2,
                                                 float* __restrict__ adj) {
  __shared__ float s_hj[64 * 17];  // stride 17: bank-conflict-free for stride-16 access
  __shared__ float s_hi[4 * 16];
  __shared__ float s_w[16];
  int tid = threadIdx.x;
  int jb = blockIdx.x * 64;
  int ib = blockIdx.y * 4;
#pragma unroll
  for (int t = tid; t < 64 * 16; t += 256)
    s_hj[(t >> 4) * 17 + (t & 15)] = hj[jb * LATD + t];
  if (tid < 64) s_hi[tid] = hi[ib * LATD + tid];
  if (tid < 16) s_w[tid] = dW2[tid];
  __syncthreads();
  int i = tid >> 6;   // 0..3 (uniform per wave -> LDS broadcast reads)
  int j = tid & 63;   // 0..63 (coalesced stores)
  const float* hr = &s_hi[i * 16];
  const float* cr = &s_hj[j * 17];
  float acc = db2[0];
#pragma unroll
  for (int l = 0; l < 16; ++l) {
    float t = hr[l] + cr[l];
    acc = fmaf(fmaxf(t, 0.0f), s_w[l], acc);
  }
  adj[(ib + i) * NN + jb + j] = 1.0f / (1.0f + expf(-acc));
}

// ---------------- dense true adjacency ----------------
__global__ void k_true_scatter(const long long* __restrict__ eit, float* __restrict__ adj_true) {
  int e = blockIdx.x * blockDim.x + threadIdx.x;
  if (e < NE) atomicAdd(&adj_true[(int)eit[e] * NN + (int)eit[NE + e]], 1.0f);
}

// ---------------- launch ----------------
extern "C" void kernel_launch(void* const* d_in, const int* in_sizes, int n_in,
                              void* d_out, int out_size, void* d_ws, size_t ws_size,
                              hipStream_t stream) {
  (void)in_sizes; (void)n_in; (void)out_size; (void)ws_size;
  const float* x    = (const float*)d_in[0];
  const float* eps  = (const float*)d_in[1];
  const float* W1   = (const float*)d_in[2];
  const float* b1   = (const float*)d_in[3];
  const float* Wmu  = (const float*)d_in[4];
  const float* bmu  = (const float*)d_in[5];
  const float* Wlv  = (const float*)d_in[6];
  const float* blv  = (const float*)d_in[7];
  const float* dW1  = (const float*)d_in[8];
  const float* db1  = (const float*)d_in[9];
  const float* dW2  = (const float*)d_in[10];
  const float* db2  = (const float*)d_in[11];
  const long long* ei  = (const long long*)d_in[12];
  const long long* eit = (const long long*)d_in[13];

  float* out      = (float*)d_out;
  float* adj_pred = out;                       // 2048*2048
  float* adj_true = out + 4194304;             // 2048*2048
  float* out_mu   = out + 8388608;             // 2048*16
  float* out_lv   = out + 8421376;             // 2048*16

  float* ws    = (float*)d_ws;
  float* deg   = ws;                 // 2048
  float* dinv  = ws + 2048;          // 2048
  float* h1    = ws + 4096;          // 65536
  float* agg1  = ws + 69632;         // 65536
  float* h     = ws + 135168;        // 65536
  float* mupre = ws + 200704;        // 32768
  float* lvpre = ws + 233472;        // 32768
  float* aggmu = ws + 266240;        // 32768
  float* agglv = ws + 299008;        // 32768 (contiguous with aggmu)
  float* z     = ws + 331776;        // 32768
  float* hi    = ws + 364544;        // 32768
  float* hj    = ws + 397312;        // 32768

  // degrees + normalization
  k_fill<<<8, 256, 0, stream>>>(deg, 1.0f, NN);            // self-loop contributes 1
  k_deg_scatter<<<NE / 256, 256, 0, stream>>>(ei, deg);
  k_dinv<<<8, 256, 0, stream>>>(deg, dinv, NN);

  // conv1: h = relu(GCN(x, W1))
  k_gemm1<<<32, 256, 0, stream>>>(x, W1, h1);
  k_fill<<<256, 256, 0, stream>>>(agg1, 0.0f, NN * HIDD);
  k_agg1<<<(NE * HIDD) / 256, 256, 0, stream>>>(ei, dinv, h1, agg1);
  k_relu1<<<(NN * HIDD) / 256, 256, 0, stream>>>(agg1, h1, dinv, b1, h);

  // conv mu / logvar (fused GEMM, fused scatter)
  k_gemm2<<<16, 256, 0, stream>>>(h, Wmu, Wlv, mupre, lvpre);
  k_fill<<<256, 256, 0, stream>>>(aggmu, 0.0f, 2 * NN * LATD);  // aggmu+agglv contiguous
  k_agg2<<<(NE * LATD) / 256, 256, 0, stream>>>(ei, dinv, mupre, lvpre, aggmu, agglv);
  k_finish2<<<(NN * LATD) / 256, 256, 0, stream>>>(aggmu, agglv, mupre, lvpre, dinv,
                                                   bmu, blv, eps, out_mu, out_lv, z);

  // decoder projections + dense pairwise decode
  k_gemm3<<<16, 256, 0, stream>>>(z, dW1, db1, hi, hj);
  dim3 dg(NN / 64, NN / 4);
  k_decoder<<<dg, 256, 0, stream>>>(hi, hj, dW2, db2, adj_pred);

  // dense true adjacency (zero + scatter with multiplicity)
  k_fill<<<NN * NN / 256, 256, 0, stream>>>(adj_true, 0.0f, NN * NN);
  k_true_scatter<<<NE / 256, 256, 0, stream>>>(eit, adj_true);
}